// PureCartesianTensorProduct_37967510897140
// MI455X (gfx1250) — compile-verified
//
#include <hip/hip_runtime.h>

#define FEAT     832
#define LSTRIDE  836      // padded LDS row stride (floats): +4 banks skew, float4 aligned
#define NPATH    15

typedef float v2f __attribute__((ext_vector_type(2)));
typedef float v8f __attribute__((ext_vector_type(8)));

struct PInfo { int L1, L2, Lout, k, eps; };

// Same order as the Python PATHS enumeration (compile-time table)
constexpr PInfo c_paths[NPATH] = {
  {0,0,0,0,0}, {0,1,1,0,0}, {0,2,2,0,0},
  {1,0,1,0,0}, {1,1,2,0,0}, {1,1,0,1,0}, {1,1,1,0,1},
  {1,2,1,1,0}, {1,2,2,0,1},
  {2,0,2,0,0}, {2,1,1,1,0}, {2,1,2,0,1},
  {2,2,2,1,0}, {2,2,0,2,0}, {2,2,1,1,1}
};

constexpr int pow3c(int n)   { return n == 0 ? 1 : (n == 1 ? 3 : 9); }
constexpr int rankOffc(int L){ return L == 0 ? 0 : (L == 1 ? 64 : 256); }

// One path's GEMM contribution for this wave's a-range; everything structural is compile-time.
template <int LOUT, int P>
__device__ __forceinline__ void do_path(
    int s, const float* An, const float* Bn, const float* w,
    int khi, int mrow, int aBeg,
    v8f& acc0, v8f& acc1, v8f& acc2, v8f& acc3)
{
  constexpr PInfo Pi = c_paths[P];
  if constexpr (Pi.Lout == LOUT) {
    constexpr int K3  = pow3c(Pi.k);
    constexpr int NT  = Pi.eps ? 2 * K3 : K3;      // #terms in U
    constexpr int dA  = pow3c(Pi.L1);
    constexpr int dB  = pow3c(Pi.L2);
    constexpr int offA = rankOffc(Pi.L1);
    constexpr int offB = rankOffc(Pi.L2);

    // term tables: scalar integer math off the (uniform) spatial-out index s
    int   ia[NT], ib[NT];
    float sgn[NT];
    if constexpr (!Pi.eps) {
      constexpr int fB3 = pow3c(Pi.L2 - Pi.k);
      const int u = s / fB3, v = s % fB3;
#pragma unroll
      for (int m = 0; m < K3; ++m) { ia[m] = u * K3 + m; ib[m] = v * K3 + m; sgn[m] = 1.f; }
    } else {
      constexpr int fB3 = pow3c(Pi.L2 - Pi.k - 1);
      const int pp = s % 3, rest = s / 3;
      const int v = rest % fB3, u = rest / fB3;
      const int a1 = (pp + 1) % 3, a2 = (pp + 2) % 3;
#pragma unroll
      for (int m = 0; m < K3; ++m) {
        ia[2*m]   = (u*3 + a1) * K3 + m; ib[2*m]   = (v*3 + a2) * K3 + m; sgn[2*m]   =  1.f;
        ia[2*m+1] = (u*3 + a2) * K3 + m; ib[2*m+1] = (v*3 + a1) * K3 + m; sgn[2*m+1] = -1.f;
      }
    }

    const float* Bn2 = Bn + offB + 2 * khi * dB;   // this lane's K-slot b-rows
    const float* Wp  = w + (P << 18) + 2 * khi;    // + c*4096 + a*64 + b0
    const float* Wr0 = Wp + (mrow +  0) * 4096;
    const float* Wr1 = Wp + (mrow + 16) * 4096;
    const float* Wr2 = Wp + (mrow + 32) * 4096;
    const float* Wr3 = Wp + (mrow + 48) * 4096;

    for (int ai = 0; ai < 16; ++ai) {              // scalar trip count: no divergent loop
      const int a = aBeg + ai;
      const float* Aa = An + offA + a * dA;
      float av[NT];
#pragma unroll
      for (int t = 0; t < NT; ++t) av[t] = sgn[t] * Aa[ia[t]];

      const int aw = a * 64;
      for (int b0 = 0; b0 < 64; b0 += 4) {
        const float* B0 = Bn2 + b0 * dB;           // b = b0 + 2*khi + {0,1}
        const float* B1 = B0 + dB;
        float u0 = 0.f, u1 = 0.f;
#pragma unroll
        for (int t = 0; t < NT; ++t) {
          u0 = fmaf(av[t], B0[ib[t]], u0);
          u1 = fmaf(av[t], B1[ib[t]], u1);
        }
        v2f aop; aop.x = u0; aop.y = u1;           // VGPR0<->K=2*khi, VGPR1<->K=2*khi+1
        v2f w0 = *(const v2f*)(Wr0 + aw + b0);     // rows 2*khi+{0,1} of the 4x16 W slice
        v2f w1 = *(const v2f*)(Wr1 + aw + b0);
        v2f w2 = *(const v2f*)(Wr2 + aw + b0);
        v2f w3 = *(const v2f*)(Wr3 + aw + b0);
        acc0 = __builtin_amdgcn_wmma_f32_16x16x4_f32(false, aop, false, w0, (short)0, acc0, false, false);
        acc1 = __builtin_amdgcn_wmma_f32_16x16x4_f32(false, aop, false, w1, (short)0, acc1, false, false);
        acc2 = __builtin_amdgcn_wmma_f32_16x16x4_f32(false, aop, false, w2, (short)0, acc2, false, false);
        acc3 = __builtin_amdgcn_wmma_f32_16x16x4_f32(false, aop, false, w3, (short)0, acc3, false, false);
      }
    }
  }
}

// Fully unrolled compile-time path chain
template <int LOUT, int P>
struct PathChain {
  static __device__ __forceinline__ void run(
      int s, const float* An, const float* Bn, const float* w,
      int khi, int mrow, int aBeg,
      v8f& a0, v8f& a1, v8f& a2, v8f& a3) {
    do_path<LOUT, P>(s, An, Bn, w, khi, mrow, aBeg, a0, a1, a2, a3);
    PathChain<LOUT, P + 1>::run(s, An, Bn, w, khi, mrow, aBeg, a0, a1, a2, a3);
  }
};
template <int LOUT>
struct PathChain<LOUT, NPATH> {
  static __device__ __forceinline__ void run(
      int, const float*, const float*, const float*,
      int, int, int, v8f&, v8f&, v8f&, v8f&) {}
};

template <int LOUT>
__global__ __launch_bounds__(256)
void tp_kernel(const float* __restrict__ x1, const float* __restrict__ x2,
               const float* __restrict__ w, float* __restrict__ out, int Btot)
{
  extern __shared__ float lds[];
  float* Af = lds;
  float* Bf = lds + 32 * LSTRIDE;

  constexpr int outOff = (LOUT == 0) ? 0 : (LOUT == 1 ? 64 : 256);
  constexpr int dOut   = pow3c(LOUT);
  const int s    = blockIdx.y;           // spatial-out component within this Lout block
  const int row0 = blockIdx.x * 32;

  // ---- stage 32 rows of x1/x2 into LDS (float4, bank-skewed rows) ----
  for (int i = threadIdx.x; i < 32 * 208; i += 256) {
    int r = i / 208, c4 = i % 208;
    int n = row0 + r;
    float4 a4 = make_float4(0.f, 0.f, 0.f, 0.f), b4 = a4;
    if (n < Btot) {
      a4 = *(const float4*)(x1 + (size_t)n * FEAT + c4 * 4);
      b4 = *(const float4*)(x2 + (size_t)n * FEAT + c4 * 4);
    }
    *(float4*)(Af + r * LSTRIDE + c4 * 4) = a4;
    *(float4*)(Bf + r * LSTRIDE + c4 * 4) = b4;
  }
  __syncthreads();

  // ---- wave mapping: 8 waves = 2 row-groups x 4 K-splits ----
  const int wave = threadIdx.x >> 5;
  const int lane = threadIdx.x & 31;
  const int mrow = lane & 15;            // M (A rows) / N (W cols) per WMMA lane layouts
  const int khi  = lane >> 4;            // selects K slots {2*khi, 2*khi+1}
  const int rg   = wave & 1;             // row group (batch rows rg*16 .. +15)
  const int ks   = wave >> 1;            // K-split: a in [ks*16, ks*16+16)

  const float* An = Af + (rg * 16 + mrow) * LSTRIDE;
  const float* Bn = Bf + (rg * 16 + mrow) * LSTRIDE;

  v8f acc0 = {}; v8f acc1 = {}; v8f acc2 = {}; v8f acc3 = {};

  PathChain<LOUT, 0>::run(s, An, Bn, w, khi, mrow, ks * 16, acc0, acc1, acc2, acc3);

  // ---- K-split reduction across the 4 waves of each row group (reuse LDS) ----
  __syncthreads();
  {
    float* R = lds;                      // 8 waves * 1024 floats = 32 KB
    const int base = wave * 1024;        // tile element index = M*64 + c
#pragma unroll
    for (int r = 0; r < 8; ++r) {
      int M = khi * 8 + r;               // C/D layout: VGPR r -> M = r (+8 for high half)
      R[base + M * 64 + (mrow +  0)] = acc0[r];
      R[base + M * 64 + (mrow + 16)] = acc1[r];
      R[base + M * 64 + (mrow + 32)] = acc2[r];
      R[base + M * 64 + (mrow + 48)] = acc3[r];
    }
  }
  __syncthreads();

  for (int e = threadIdx.x; e < 2048; e += 256) {
    int erg = e >> 10, wi = e & 1023;
    float sum = 0.f;
#pragma unroll
    for (int j = 0; j < 4; ++j) sum += lds[(erg + 2 * j) * 1024 + wi];
    int n = row0 + erg * 16 + (wi >> 6);
    int c = wi & 63;
    if (n < Btot) out[(size_t)n * FEAT + outOff + c * dOut + s] = sum;
  }
}

extern "C" void kernel_launch(void* const* d_in, const int* in_sizes, int n_in,
                              void* d_out, int out_size, void* d_ws, size_t ws_size,
                              hipStream_t stream)
{
  (void)n_in; (void)out_size; (void)d_ws; (void)ws_size;
  const float* x1 = (const float*)d_in[0];
  const float* x2 = (const float*)d_in[1];
  const float* w  = (const float*)d_in[2];
  float* out = (float*)d_out;

  const int Btot  = in_sizes[0] / FEAT;
  const int tiles = (Btot + 31) / 32;
  const dim3 block(256, 1, 1);
  const size_t shmem = (size_t)2 * 32 * LSTRIDE * sizeof(float);  // 214 KB of 320 KB WGP LDS

  hipLaunchKernelGGL(HIP_KERNEL_NAME(tp_kernel<0>), dim3(tiles, 1, 1), block, shmem, stream, x1, x2, w, out, Btot);
  hipLaunchKernelGGL(HIP_KERNEL_NAME(tp_kernel<1>), dim3(tiles, 3, 1), block, shmem, stream, x1, x2, w, out, Btot);
  hipLaunchKernelGGL(HIP_KERNEL_NAME(tp_kernel<2>), dim3(tiles, 9, 1), block, shmem, stream, x1, x2, w, out, Btot);
}